// NormalAngleShader_26628797235878
// MI455X (gfx1250) — compile-verified
//
#include <hip/hip_runtime.h>

// ---------------------------------------------------------------------------
// NormalAngleShader for MI455X (gfx1250, wave32).
// Bandwidth-bound gather kernel: ~300MB traffic -> ~13us roofline @ 23.3TB/s.
//   Kernel A: pack per-face {v0,v1,v2,normal} into 48B records in d_ws
//             (9.6MB, L2-resident on the 192MB L2) -> per-hit gather becomes
//             3 aligned b128 loads with no dependent index chain.
//   Kernel B: 2D grid (HW-blocks x N). One thread per pixel (K=3 hits).
//             Streaming AoS inputs (bary 36B/pixel, face ids 12B/pixel)
//             staged to LDS via CDNA5 async global->LDS copies
//             (global_load_async_to_lds_b128 + s_wait_asynccnt), camera
//             origin fetched via uniform scalar loads overlapping the wait,
//             outputs stored coalesced per [K,N,H,W] plane.
// ---------------------------------------------------------------------------

#define TPB 256

__device__ __forceinline__ void async_copy_b128(unsigned lds_byte_addr, const void* gaddr) {
  // CDNA5 async global->LDS copy, 16B per active lane, ASYNCcnt-tracked.
  asm volatile("global_load_async_to_lds_b128 %0, %1, off"
               :: "v"(lds_byte_addr), "v"((unsigned long long)gaddr)
               : "memory");
}

__device__ __forceinline__ void wait_async0() {
  asm volatile("s_wait_asynccnt 0" ::: "memory");
}

// ---------------- Kernel A: build packed face records ----------------------
// rec[f] = { float4(v0.xyz, n.x), float4(v1.xyz, n.y), float4(v2.xyz, n.z) }
__global__ __launch_bounds__(TPB)
void build_face_records(const float* __restrict__ verts,
                        const int*   __restrict__ faces,
                        float4*      __restrict__ rec,
                        int F)
{
  int f = blockIdx.x * TPB + threadIdx.x;
  if (f >= F) return;
  int i0 = faces[3 * f + 0];
  int i1 = faces[3 * f + 1];
  int i2 = faces[3 * f + 2];
  float ax = verts[3 * i0 + 0], ay = verts[3 * i0 + 1], az = verts[3 * i0 + 2];
  float bx = verts[3 * i1 + 0], by = verts[3 * i1 + 1], bz = verts[3 * i1 + 2];
  float cx = verts[3 * i2 + 0], cy = verts[3 * i2 + 1], cz = verts[3 * i2 + 2];
  float e1x = bx - ax, e1y = by - ay, e1z = bz - az;
  float e2x = cx - ax, e2y = cy - ay, e2z = cz - az;
  float nx = e1y * e2z - e1z * e2y;
  float ny = e1z * e2x - e1x * e2z;
  float nz = e1x * e2y - e1y * e2x;
  float len = sqrtf(nx * nx + ny * ny + nz * nz);
  float inv = 1.0f / fmaxf(len, 1e-12f);   // matches F.normalize eps semantics
  nx *= inv; ny *= inv; nz *= inv;
  rec[3 * (size_t)f + 0] = make_float4(ax, ay, az, nx);
  rec[3 * (size_t)f + 1] = make_float4(bx, by, bz, ny);
  rec[3 * (size_t)f + 2] = make_float4(cx, cy, cz, nz);
}

// ---------------- Kernel B: per-pixel shading ------------------------------
// grid = (ceil(HW/TPB), N); n = blockIdx.y is wave-uniform -> no div/mod,
// scalar camera loads.
template <bool PACKED>
__global__ __launch_bounds__(TPB)
void shade(const int*    __restrict__ p2f,
           const float*  __restrict__ bary,
           const float4* __restrict__ rec,     // packed path
           const float*  __restrict__ verts,   // fallback path
           const int*    __restrict__ faces,   // fallback path
           const float*  __restrict__ cam,
           float*        __restrict__ out,
           int HW, int N)
{
  __shared__ float sb[TPB * 9];   // 9216 B: bary for this block's pixels
  __shared__ int   sf[TPB * 3];   // 3072 B: face ids for this block's pixels

  const int tid = threadIdx.x;
  const int n   = blockIdx.y;                       // uniform per block
  const int hw0 = blockIdx.x * TPB;                 // first hw of this block
  const long long blockStart = (long long)n * HW + hw0;  // first pixel id

  int pixLeft = HW - hw0;                           // tail guard (per image)
  if (pixLeft > TPB) pixLeft = TPB;
  const unsigned baryBytes = (unsigned)(pixLeft * 36);
  const unsigned p2fBytes  = (unsigned)(pixLeft * 12);

  const char* gb = (const char*)(bary + blockStart * 9);
  const char* gf = (const char*)(p2f  + blockStart * 3);
  // Low 32 bits of a flat LDS pointer are the LDS byte address (aperture tag
  // lives entirely in bits [63:32] on CDNA5).
  const unsigned lb = (unsigned)(unsigned long long)(const void*)sb;
  const unsigned lf = (unsigned)(unsigned long long)(const void*)sf;

  // Stage bary (3 x 4KB rounds of 16B/lane) + face ids (1 round, 192 lanes).
  const unsigned o0 = (unsigned)tid * 16u;
  if (o0 < baryBytes)          async_copy_b128(lb + o0,         gb + o0);
  if (o0 + 4096u < baryBytes)  async_copy_b128(lb + o0 + 4096u, gb + o0 + 4096u);
  if (o0 + 8192u < baryBytes)  async_copy_b128(lb + o0 + 8192u, gb + o0 + 8192u);
  if (o0 < p2fBytes)           async_copy_b128(lf + o0,         gf + o0);

  // Uniform camera fetch overlaps the async staging (scalar loads).
  const float camx = cam[3 * n + 0];
  const float camy = cam[3 * n + 1];
  const float camz = cam[3 * n + 2];

  wait_async0();
  __syncthreads();

  const int hw = hw0 + tid;
  if (hw >= HW) return;

  const float* b  = &sb[tid * 9];
  const int*   fi = &sf[tid * 3];

#pragma unroll
  for (int k = 0; k < 3; ++k) {
    const int fc = fi[k];
    float v0x, v0y, v0z, v1x, v1y, v1z, v2x, v2y, v2z, nx, ny, nz;
    if (PACKED) {
      const float4* r = rec + (size_t)fc * 3;
      float4 r0 = r[0], r1 = r[1], r2 = r[2];
      v0x = r0.x; v0y = r0.y; v0z = r0.z; nx = r0.w;
      v1x = r1.x; v1y = r1.y; v1z = r1.z; ny = r1.w;
      v2x = r2.x; v2y = r2.y; v2z = r2.z; nz = r2.w;
    } else {
      const int i0 = faces[3 * fc + 0];
      const int i1 = faces[3 * fc + 1];
      const int i2 = faces[3 * fc + 2];
      v0x = verts[3 * i0 + 0]; v0y = verts[3 * i0 + 1]; v0z = verts[3 * i0 + 2];
      v1x = verts[3 * i1 + 0]; v1y = verts[3 * i1 + 1]; v1z = verts[3 * i1 + 2];
      v2x = verts[3 * i2 + 0]; v2y = verts[3 * i2 + 1]; v2z = verts[3 * i2 + 2];
      const float e1x = v1x - v0x, e1y = v1y - v0y, e1z = v1z - v0z;
      const float e2x = v2x - v0x, e2y = v2y - v0y, e2z = v2z - v0z;
      nx = e1y * e2z - e1z * e2y;
      ny = e1z * e2x - e1x * e2z;
      nz = e1x * e2y - e1y * e2x;
      const float inv = 1.0f / fmaxf(sqrtf(nx * nx + ny * ny + nz * nz), 1e-12f);
      nx *= inv; ny *= inv; nz *= inv;
    }
    const float b0 = b[3 * k + 0], b1 = b[3 * k + 1], b2 = b[3 * k + 2];
    const float px = b0 * v0x + b1 * v1x + b2 * v2x;
    const float py = b0 * v0y + b1 * v1y + b2 * v2y;
    const float pz = b0 * v0z + b1 * v1z + b2 * v2z;
    const float vx = px - camx, vy = py - camy, vz = pz - camz;
    const float inv = 1.0f / fmaxf(sqrtf(vx * vx + vy * vy + vz * vz), 1e-12f);
    const float d = (nx * vx + ny * vy + nz * vz) * inv;
    out[((size_t)(k * N + n)) * (size_t)HW + hw] = d;  // [K,N,H,W,1] layout
  }
}

// ---------------- Host-side launcher ---------------------------------------
extern "C" void kernel_launch(void* const* d_in, const int* in_sizes, int n_in,
                              void* d_out, int out_size, void* d_ws, size_t ws_size,
                              hipStream_t stream) {
  const int*   p2f   = (const int*)  d_in[0];  // [N,H,W,K] int32
  const float* bary  = (const float*)d_in[1];  // [N,H,W,K,3] f32
  const float* verts = (const float*)d_in[2];  // [V,3] f32
  const int*   faces = (const int*)  d_in[3];  // [F,3] int32
  const float* cam   = (const float*)d_in[4];  // [N,3] f32
  float* out = (float*)d_out;

  const int K = 3;                                    // rasterizer faces-per-pixel
  const int N = in_sizes[4] / 3;
  const long long total = (long long)in_sizes[0];     // N*H*W*K
  const int HW = (int)(total / ((long long)N * K));
  const int F  = in_sizes[3] / 3;

  const dim3 grid((unsigned)((HW + TPB - 1) / TPB), (unsigned)N, 1);
  const bool packed = (ws_size >= (size_t)F * 48u);

  if (packed) {
    build_face_records<<<(F + TPB - 1) / TPB, TPB, 0, stream>>>(
        verts, faces, (float4*)d_ws, F);
    shade<true><<<grid, TPB, 0, stream>>>(
        p2f, bary, (const float4*)d_ws, verts, faces, cam, out, HW, N);
  } else {
    shade<false><<<grid, TPB, 0, stream>>>(
        p2f, bary, (const float4*)nullptr, verts, faces, cam, out, HW, N);
  }
}